// SdfModel_36215164239982
// MI455X (gfx1250) — compile-verified
//
#include <hip/hip_runtime.h>

#define JDIM 512
#define DDIM 20
#define ITERS 8
#define EPSV 1e-12f

typedef __attribute__((ext_vector_type(2))) float v2f;
typedef __attribute__((ext_vector_type(8))) float v8f;
typedef __attribute__((ext_vector_type(4))) unsigned int u32x4;
typedef __attribute__((ext_vector_type(8))) int i32x8;
typedef __attribute__((ext_vector_type(4))) int i32x4;

#if __has_builtin(__builtin_amdgcn_tensor_load_to_lds) && \
    __has_builtin(__builtin_amdgcn_s_wait_tensorcnt) && \
    __has_builtin(__builtin_amdgcn_groupstaticsize)
#define USE_TDM 1
#else
#define USE_TDM 0
#endif

// ---- LDS layout (float offsets) ----
#define OFF_BUF0   0        // 10240 : nodes[b], later overwritten with nodes_rep
#define OFF_ADDRM  10240    // 10240 : nodes_addr (L2-normalized)
#define OFF_WREP   20480    // 640   : W_rep padded to 20x32 (cols >=20 zero)
#define OFF_WADDR  21120    // 640   : W_addr padded to 20x32
#define OFF_WATT   21760    // 3600  : W_att 60x60 row-major
#define OFF_WF1    25360    // 400   : W_f1 20x20
#define OFF_WF2    25760    // 40    : W_f2 20x2
#define OFF_BREP   25800    // 20
#define OFF_BADDR  25820    // 20
#define OFF_BATT   25840    // 60
#define OFF_BF1    25900    // 20
#define OFF_BF2    25920    // 4 (2 used)
#define OFF_DP     25924    // 512
#define OFF_CAT    26436    // 64 : [in_src(20), addr(20), hidden(20)]
#define OFF_ATTOUT 26500    // 64
#define OFF_RED    26564    // 160 : 8x20 in_src partials
#define OFF_WRED   26724    // 8   : per-wave dp^2 partials
#define OFF_INAGG  26732    // 20
#define OFF_HBUF   26772    // 20
#define LDS_FLOATS 26792

__global__ __launch_bounds__(256) void sdf_fused_kernel(
    const float* __restrict__ nodes,
    const float* __restrict__ W_rep,  const float* __restrict__ b_rep,
    const float* __restrict__ W_addr, const float* __restrict__ b_addr,
    const float* __restrict__ W_att,  const float* __restrict__ b_att,
    const float* __restrict__ W_f1,   const float* __restrict__ b_f1,
    const float* __restrict__ W_f2,   const float* __restrict__ b_f2,
    float* __restrict__ out)
{
    extern __shared__ float sm[];
    const int tid  = threadIdx.x;
    const int b    = blockIdx.x;
    const int wave = tid >> 5;
    const int lane = tid & 31;
    const int lh   = lane >> 4;   // half-wave select (K/M high half)
    const int ll   = lane & 15;   // lane within half

    float* s_buf0  = sm + OFF_BUF0;
    float* s_addrm = sm + OFF_ADDRM;
    float* s_wrep  = sm + OFF_WREP;
    float* s_waddr = sm + OFF_WADDR;
    float* s_watt  = sm + OFF_WATT;
    float* s_wf1   = sm + OFF_WF1;
    float* s_wf2   = sm + OFF_WF2;
    float* s_brep  = sm + OFF_BREP;
    float* s_baddr = sm + OFF_BADDR;
    float* s_batt  = sm + OFF_BATT;
    float* s_bf1   = sm + OFF_BF1;
    float* s_bf2   = sm + OFF_BF2;
    float* s_dp    = sm + OFF_DP;
    float* s_cat   = sm + OFF_CAT;
    float* s_atto  = sm + OFF_ATTOUT;
    float* s_red   = sm + OFF_RED;
    float* s_wred  = sm + OFF_WRED;
    float* s_inagg = sm + OFF_INAGG;
    float* s_hbuf  = sm + OFF_HBUF;

    // ---- stage nodes[b] (512x20 f32 = 40KB contiguous) into LDS ----
#if USE_TDM
    // Tensor Data Mover: one DMA issued by wave 0 only (TDM ignores EXEC).
    if (wave == 0) {
        const unsigned lds_addr = (unsigned)__builtin_amdgcn_groupstaticsize()
                                  + (unsigned)(OFF_BUF0 * sizeof(float));
        const unsigned long long ga =
            (unsigned long long)(const void*)(nodes + (size_t)b * (JDIM * DDIM));
        // D# group 0: count=1 | lds_addr | global_addr[56:0] | type=2
        u32x4 g0 = { 1u,
                     lds_addr,
                     (unsigned)(ga & 0xffffffffu),
                     (unsigned)((ga >> 32) & 0x01ffffffu) | 0x80000000u };
        // D# group 1: 1-D tile, data_size=4B, dim0 = tile0 = 10240 elements
        i32x8 g1 = { (int)0x00020000u,           // data_size=2 (4B), mask=0
                     (int)(10240u << 16),        // tensor_dim0[15:0] @ bits 63:48
                     (int)(1u << 16),            // tensor_dim0 hi=0, tensor_dim1=1
                     (int)(10240u << 16),        // tensor_dim1 hi=0, tile_dim0=10240
                     1,                          // tile_dim1=1, tile_dim2=0
                     10240,                      // tensor_dim0_stride lo
                     0, 0 };
        i32x4 gz = { 0, 0, 0, 0 };
#if defined(__clang_major__) && (__clang_major__ >= 23)
        i32x8 gz8 = { 0, 0, 0, 0, 0, 0, 0, 0 };
        __builtin_amdgcn_tensor_load_to_lds(g0, g1, gz, gz, gz8, 0);
#else
        __builtin_amdgcn_tensor_load_to_lds(g0, g1, gz, gz, 0);
#endif
        __builtin_amdgcn_s_wait_tensorcnt(0);
    }
#else
    {
        const float4* src = (const float4*)(nodes + (size_t)b * (JDIM * DDIM));
        float4* dst = (float4*)s_buf0;
        for (int i = tid; i < (JDIM * DDIM) / 4; i += 256) dst[i] = src[i];
    }
#endif

    // ---- load weights / biases into LDS (overlaps with TDM on other waves) ----
    for (int i = tid; i < 640; i += 256) {
        int k = i >> 5, n = i & 31;
        float wr = 0.f, wa = 0.f;
        if (n < DDIM) { wr = W_rep[k * DDIM + n]; wa = W_addr[k * DDIM + n]; }
        s_wrep[i] = wr; s_waddr[i] = wa;
    }
    for (int i = tid; i < 3600; i += 256) s_watt[i] = W_att[i];
    for (int i = tid; i < 400;  i += 256) s_wf1[i]  = W_f1[i];
    if (tid < 40) s_wf2[tid] = W_f2[tid];
    if (tid < 20) { s_brep[tid] = b_rep[tid]; s_baddr[tid] = b_addr[tid]; s_bf1[tid] = b_f1[tid]; }
    if (tid < 60) s_batt[tid] = b_att[tid];
    if (tid < 2)  s_bf2[tid]  = b_f2[tid];
    if (tid < 64) s_cat[tid]  = 0.f;   // addr = hidden = 0 initial carry
    __syncthreads();

    // ---- hoist per-lane bias values for GEMM writeback (loop-invariant) ----
    const bool  n1v = (16 + ll) < DDIM;
    const float br0 = s_brep[ll];
    const float ba0 = s_baddr[ll];
    const float br1 = s_brep[n1v ? (16 + ll) : 0];
    const float ba1 = s_baddr[n1v ? (16 + ll) : 0];

    // ---- preload B fragments (weights) for both GEMMs, all k-steps/N-tiles ----
    // B 4x16 f32 layout: VGPR0 = W[k0+2*lh][n], VGPR1 = W[k0+2*lh+1][n], n = n0+ll
    v2f fbr[5][2], fba[5][2];
#pragma unroll
    for (int kt = 0; kt < 5; ++kt) {
        int r = 4 * kt + 2 * lh;
#pragma unroll
        for (int nt = 0; nt < 2; ++nt) {
            int c = 16 * nt + ll;
            v2f t0 = { s_wrep[r * 32 + c],  s_wrep[(r + 1) * 32 + c] };
            v2f t1 = { s_waddr[r * 32 + c], s_waddr[(r + 1) * 32 + c] };
            fbr[kt][nt] = t0;
            fba[kt][nt] = t1;
        }
    }

    // ---- WMMA GEMMs: rep = nodes@W_rep + b_rep ; addrm = nodes@W_addr + b_addr
    // 32 M-tiles of 16 rows; each wave owns 4 tiles. rep overwrites s_buf0 in place
    // (safe: only rows already consumed by this wave's A-fragment loads).
    for (int mt = wave; mt < 32; mt += 8) {
        const int row = mt * 16 + ll;
        v2f fa[5];
#pragma unroll
        for (int kt = 0; kt < 5; ++kt)
            fa[kt] = *(const v2f*)&s_buf0[row * DDIM + 4 * kt + 2 * lh];

        v8f aR0 = {}, aR1 = {}, aA0 = {}, aA1 = {};
#pragma unroll
        for (int kt = 0; kt < 5; ++kt) {
            aR0 = __builtin_amdgcn_wmma_f32_16x16x4_f32(false, fa[kt], false, fbr[kt][0], (short)0, aR0, false, false);
            aR1 = __builtin_amdgcn_wmma_f32_16x16x4_f32(false, fa[kt], false, fbr[kt][1], (short)0, aR1, false, false);
            aA0 = __builtin_amdgcn_wmma_f32_16x16x4_f32(false, fa[kt], false, fba[kt][0], (short)0, aA0, false, false);
            aA1 = __builtin_amdgcn_wmma_f32_16x16x4_f32(false, fa[kt], false, fba[kt][1], (short)0, aA1, false, false);
        }
        // D layout: VGPR v, lanes 0-15 -> (M = v,   N = ll)
        //                   lanes 16-31-> (M = v+8, N = ll)
#pragma unroll
        for (int v = 0; v < 8; ++v) {
            int m = mt * 16 + v + 8 * lh;
            s_buf0 [m * DDIM + ll] = aR0[v] + br0;
            s_addrm[m * DDIM + ll] = aA0[v] + ba0;
            if (n1v) {
                s_buf0 [m * DDIM + 16 + ll] = aR1[v] + br1;
                s_addrm[m * DDIM + 16 + ll] = aA1[v] + ba1;
            }
        }
    }
    __syncthreads();

    // ---- L2-normalize each row of addrm ----
    for (int r = tid; r < JDIM; r += 256) {
        float s = 0.f;
#pragma unroll
        for (int d = 0; d < DDIM; ++d) { float v = s_addrm[r * DDIM + d]; s += v * v; }
        float inv = 1.f / fmaxf(sqrtf(s), EPSV);
#pragma unroll
        for (int d = 0; d < DDIM; ++d) s_addrm[r * DDIM + d] *= inv;
    }
    __syncthreads();

    // ---- 8 recurrent iterations ----
    for (int it = 0; it < ITERS; ++it) {
        // current addr vector -> registers (single LDS broadcast per element)
        float av[DDIM];
#pragma unroll
        for (int d = 0; d < DDIM; ++d) av[d] = s_cat[20 + d];

        // dp_j = exp(addrm[j] . addr), accumulate sum of squares
        float ssq = 0.f;
        for (int r = tid; r < JDIM; r += 256) {
            float s = 0.f;
#pragma unroll
            for (int d = 0; d < DDIM; ++d) s += s_addrm[r * DDIM + d] * av[d];
            float e = __expf(s);
            s_dp[r] = e;
            ssq += e * e;
        }
#pragma unroll
        for (int off = 16; off >= 1; off >>= 1) ssq += __shfl_xor(ssq, off, 32);
        if (lane == 0) s_wred[wave] = ssq;
        __syncthreads();
        float tot = 0.f;
#pragma unroll
        for (int w = 0; w < 8; ++w) tot += s_wred[w];
        const float rnorm = 1.f / fmaxf(sqrtf(tot), EPSV);   // fold dp L2-norm into in_src

        // in_src[d] = rnorm * sum_j rep[j][d] * dp[j]  (160 threads: 20 d x 8 chunks of 64)
        if (tid < 160) {
            int d = tid % 20, c = tid / 20;
            float s = 0.f;
            int j0 = c * 64;
            for (int j = j0; j < j0 + 64; ++j) s += s_buf0[j * DDIM + d] * s_dp[j];
            s_red[c * 20 + d] = s;
        }
        __syncthreads();
        if (tid < 20) {
            float s = 0.f;
#pragma unroll
            for (int c = 0; c < 8; ++c) s += s_red[c * 20 + tid];
            s_cat[tid] = s * rnorm;
        }
        __syncthreads();

        // attention block: attout = cat(60) @ W_att(60x60) + b_att
        if (tid < 60) {
            float s = s_batt[tid];
#pragma unroll 4
            for (int i = 0; i < 60; ++i) s += s_cat[i] * s_watt[i * 60 + tid];
            s_atto[tid] = s;
        }
        __syncthreads();
        if (tid < 20)      s_inagg[tid] = s_atto[tid];               // in_agg
        else if (tid < 40) s_cat[tid]   = s_atto[tid];               // new addr
        else if (tid < 60) s_cat[tid]   = fmaxf(s_atto[tid], 0.f);   // new hidden (relu)
        __syncthreads();
    }

    // ---- head: relu(in_agg @ W_f1 + b_f1) @ W_f2 + b_f2 ----
    if (tid < 20) {
        float s = s_bf1[tid];
#pragma unroll
        for (int k = 0; k < 20; ++k) s += s_inagg[k] * s_wf1[k * DDIM + tid];
        s_hbuf[tid] = fmaxf(s, 0.f);
    }
    __syncthreads();
    if (tid < 2) {
        float s = s_bf2[tid];
#pragma unroll
        for (int d = 0; d < DDIM; ++d) s += s_hbuf[d] * s_wf2[d * 2 + tid];
        out[b * 2 + tid] = s;
    }
}

extern "C" void kernel_launch(void* const* d_in, const int* in_sizes, int n_in,
                              void* d_out, int out_size, void* d_ws, size_t ws_size,
                              hipStream_t stream) {
    const float* nodes  = (const float*)d_in[0];
    const float* W_rep  = (const float*)d_in[1];
    const float* b_rep  = (const float*)d_in[2];
    const float* W_addr = (const float*)d_in[3];
    const float* b_addr = (const float*)d_in[4];
    const float* W_att  = (const float*)d_in[5];
    const float* b_att  = (const float*)d_in[6];
    const float* W_f1   = (const float*)d_in[7];
    const float* b_f1   = (const float*)d_in[8];
    const float* W_f2   = (const float*)d_in[9];
    const float* b_f2   = (const float*)d_in[10];
    float* out = (float*)d_out;

    const int B = in_sizes[0] / (JDIM * DDIM);   // 4096
    const size_t shbytes = (size_t)LDS_FLOATS * sizeof(float);

    sdf_fused_kernel<<<dim3(B), dim3(256), shbytes, stream>>>(
        nodes, W_rep, b_rep, W_addr, b_addr, W_att, b_att, W_f1, b_f1, W_f2, b_f2, out);
}